// DeltaOrderLoss_81862076661833
// MI455X (gfx1250) — compile-verified
//
#include <hip/hip_runtime.h>
#include <math.h>

typedef __attribute__((ext_vector_type(2))) float v2f;
typedef __attribute__((ext_vector_type(8))) float v8f;

#define NN 256
#define DD 128
#define NM1 255

// ws layout (floats):
//   [0, 65536)     : z distances [256][256]
//   [65536, 65792) : row norms [256]
//   [65792, 66048) : per-row partial sums [256]

// f[r] = features[r % 128][r / 128][:]  (concat of view 0 rows, then view 1 rows)
__device__ __forceinline__ const float* frow(const float* feat, int r) {
    int base = ((r & 127) << 1) + (r >> 7);
    return feat + base * DD;
}

__global__ void norms_kernel(const float* __restrict__ feat, float* __restrict__ norms) {
    int r = blockIdx.x * blockDim.x + threadIdx.x;
    if (r >= NN) return;
    const float* fp = frow(feat, r);
    float s = 0.f;
#pragma unroll 8
    for (int d = 0; d < DD; ++d) s += fp[d] * fp[d];
    norms[r] = s;
}

// One wave per 16x16 tile of the Gram matrix G = F * F^T (fp32 WMMA),
// then z[m][n] = sqrt(max(norm_m + norm_n - 2 G[m][n], 0)).
__global__ void gram_dist_kernel(const float* __restrict__ feat,
                                 const float* __restrict__ norms,
                                 float* __restrict__ z) {
    int wave = threadIdx.x >> 5;
    int lane = threadIdx.x & 31;
    int tile = blockIdx.x * (blockDim.x >> 5) + wave;   // 0..255
    int mBase = (tile >> 4) << 4;
    int nBase = (tile & 15) << 4;

    int halfSel = lane >> 4;        // 0 -> K pair {k0,k0+1}, 1 -> {k0+2,k0+3}
    int rowLane = lane & 15;

    const float* aRow = frow(feat, mBase + rowLane);    // A: 16x4 tile rows
    const float* bRow = frow(feat, nBase + rowLane);    // B: 4x16 tile = F rows at nBase

    v8f c = {};
#pragma unroll 4
    for (int k0 = 0; k0 < DD; k0 += 4) {
        int kk = k0 + (halfSel << 1);
        v2f a, b;
        a.x = aRow[kk]; a.y = aRow[kk + 1];
        b.x = bRow[kk]; b.y = bRow[kk + 1];
        // D = A*B + C, fp32 throughout (matches reference precision)
        c = __builtin_amdgcn_wmma_f32_16x16x4_f32(false, a, false, b,
                                                  (short)0, c, false, false);
    }

    int nn = nBase + rowLane;
    float normN = norms[nn];
#pragma unroll
    for (int v = 0; v < 8; ++v) {
        int mm = mBase + v + (halfSel << 3);
        float sq = norms[mm] + normN - 2.0f * c[v];
        z[mm * NN + nn] = (sq > 0.f) ? sqrtf(sq) : 0.f;
    }
}

// One block per row i: gather off-diagonal z and |label diff|, dense-rank via
// 50-bit presence mask + popcount, then reduce over all (j,k) pairs.
__global__ void row_loss_kernel(const float* __restrict__ z,
                                const int* __restrict__ labels,
                                float* __restrict__ partial) {
    __shared__ float s_z[NM1];
    __shared__ float s_rank[NM1];
    __shared__ int   s_y[NM1];
    __shared__ unsigned s_mask[2];
    __shared__ float red[256];

    int i = blockIdx.x;
    int t = threadIdx.x;
    if (t < 2) s_mask[t] = 0u;
    __syncthreads();

    int labI = labels[i & 127];
    if (t < NM1) {
        int col = (t < i) ? t : t + 1;            // off-diagonal column index
        int y = labI - labels[col & 127];
        y = (y < 0) ? -y : y;                     // |y| in [0, 49]
        s_y[t] = y;
        atomicOr(&s_mask[y >> 5], 1u << (y & 31));
        s_z[t] = z[i * NN + col];
    }
    __syncthreads();

    unsigned m0 = s_mask[0], m1 = s_mask[1];
    if (t < NM1) {
        int y = s_y[t];
        int r;
        if (y < 32) r = __popc(m0 & ((1u << y) - 1u));
        else        r = __popc(m0) + __popc(m1 & ((1u << (y - 32)) - 1u));
        s_rank[t] = (float)r;                     // dense rank of y in this row
    }
    __syncthreads();

    float acc = 0.f;
    for (int idx = t; idx < NM1 * NM1; idx += 256) {
        int j = idx / NM1;
        int k = idx - j * NM1;
        int aj = s_y[j], ak = s_y[k];
        if (ak != aj) {                           // neg_mask; equal -> term is 0 anyway
            float sgn = (ak > aj) ? 1.f : -1.f;   // sign(y_abs_k - y_abs_j)
            float fdd = sgn * (s_z[k] - s_z[j]);  // flipped distance diff
            float margin = fabsf(s_rank[k] - s_rank[j]) * 10.0f;  // /DELTA
            if (margin > fdd) acc += margin - fdd;
        }
    }
    red[t] = acc;
    __syncthreads();
#pragma unroll
    for (int s = 128; s > 0; s >>= 1) {
        if (t < s) red[t] += red[t + s];
        __syncthreads();
    }
    if (t == 0) partial[i] = red[0];
}

__global__ void final_kernel(const float* __restrict__ partial, float* __restrict__ out) {
    __shared__ float red[256];
    int t = threadIdx.x;
    red[t] = partial[t];
    __syncthreads();
#pragma unroll
    for (int s = 128; s > 0; s >>= 1) {
        if (t < s) red[t] += red[t + s];
        __syncthreads();
    }
    if (t == 0) out[0] = -(red[0] / 16646400.0f);  // mean over 256*255*255, negated
}

extern "C" void kernel_launch(void* const* d_in, const int* in_sizes, int n_in,
                              void* d_out, int out_size, void* d_ws, size_t ws_size,
                              hipStream_t stream) {
    const float* feat   = (const float*)d_in[0];   // [128, 2, 128] f32
    const int*   labels = (const int*)d_in[1];     // [128, 1] int

    float* ws      = (float*)d_ws;
    float* zbuf    = ws;                 // 65536 floats
    float* norms   = ws + NN * NN;       // 256 floats
    float* partial = norms + NN;         // 256 floats
    float* out     = (float*)d_out;

    norms_kernel<<<1, 256, 0, stream>>>(feat, norms);
    gram_dist_kernel<<<64, 128, 0, stream>>>(feat, norms, zbuf);   // 256 waves, 1 tile each
    row_loss_kernel<<<NN, 256, 0, stream>>>(zbuf, labels, partial);
    final_kernel<<<1, 256, 0, stream>>>(partial, out);
}